// WavePinn_76647986364578
// MI455X (gfx1250) — compile-verified
//
#include <hip/hip_runtime.h>

typedef __attribute__((ext_vector_type(2))) float v2f;
typedef __attribute__((ext_vector_type(8))) float v8f;

#define HID    256
#define TILE   16
#define NPTS   131072
#define NTILES (NPTS / TILE)

// NS  = number of propagation streams (5: h,vt,vx,at,ax | 2: h,vt | 1: h)
// TPW = tiles (of 16 points) per workgroup
// WPT = waves cooperating on one tile (8 waves per 256-thread block total)
// PHASE: 0 = equation (pde residual), 1 = init (u, du/dt), 2 = boundary (u)
//
// Stream-interleaved, pair-interleaved transposed activation layout:
//   element (stream s, feature f, batch n) of a buffer lives at
//       ((f/2)*NS + s)*32 + 2n + (f&1)
// -> a lane's B-fragment (K,K+1) is one contiguous ds_load_b64 (banks 2n,2n+1,
//    conflict free); the NS fragments of one K-step are only 128B apart, so
//    they encode as DS immediate offsets / fuse into ds_load_2addr_b64 with a
//    single shared base VGPR (no per-load v_add address math).
template<int NS, int TPW, int WPT, int PHASE>
__global__ __launch_bounds__(256, 1)
void pinn_kernel(const float* __restrict__ W0, const float* __restrict__ b0,
                 const float* __restrict__ W1, const float* __restrict__ b1,
                 const float* __restrict__ W2, const float* __restrict__ b2,
                 const float* __restrict__ W3, const float* __restrict__ b3,
                 const float* __restrict__ W4, const float* __restrict__ b4,
                 const float* __restrict__ tx, float* __restrict__ out)
{
    // TPW * 2 buffers * NS * 16KB   (equation phase: exactly 320KB)
    __shared__ float smem[TPW * NS * 2 * (HID * TILE)];

    const int tid  = threadIdx.x;
    const int lane = tid & 31;
    const int wv   = tid >> 5;        // wave id 0..7
    const int nn   = lane & 15;       // batch element within tile (WMMA N)
    const int half = lane >> 4;       // lane half (selects K pair / M+8)
    const int ts   = wv / WPT;        // tile slot within workgroup
    const int sub  = wv % WPT;        // sub-wave within tile
    const int tile = blockIdx.x * TPW + ts;

    // base of one (tile slot, ping-pong) buffer; holds NS interleaved streams
    auto buf = [&](int pp) -> float* {
        return smem + (ts * 2 + pp) * (NS * HID * TILE);
    };
    // offset of (stream, feature, batch) inside a buffer
    auto eoff = [&](int s2, int f, int n) -> int {
        return (((f >> 1) * NS) + s2) * 32 + (n << 1) + (f & 1);
    };

    // per-lane point coordinates (lanes 16-31 mirror 0-15)
    const float pt = tx[(tile * TILE + nn) * 2 + 0];
    const float px = tx[(tile * TILE + nn) * 2 + 1];

    // ---------------- layer 0:  2 -> 256  (tiny K, plain VALU) -----------
    {
        float* b = buf(0);
        const int FPW  = HID / WPT;               // features per wave
        const int base = sub * FPW + half * (FPW / 2);
        for (int f = base; f < base + FPW / 2; f += 2) {
            v2f w0t = *(const v2f*)(W0 + f);
            v2f w0x = *(const v2f*)(W0 + HID + f);
            v2f bb  = *(const v2f*)(b0 + f);
            v2f h, s;
#pragma unroll
            for (int q = 0; q < 2; ++q) {
                const float z = w0t[q] * pt + w0x[q] * px + bb[q];
                h[q] = tanhf(z);
                s[q] = 1.0f - h[q] * h[q];
            }
            *(v2f*)(b + eoff(0, f, nn)) = h;
            if (NS >= 2) {
                v2f v; v.x = s.x * w0t.x; v.y = s.y * w0t.y;
                *(v2f*)(b + eoff(1, f, nn)) = v;
            }
            if (NS == 5) {
                v2f v, at, ax;
                v.x  = s.x * w0x.x;                 v.y  = s.y * w0x.y;
                at.x = -2.0f * h.x * s.x * w0t.x * w0t.x;
                at.y = -2.0f * h.y * s.y * w0t.y * w0t.y;
                ax.x = -2.0f * h.x * s.x * w0x.x * w0x.x;
                ax.y = -2.0f * h.y * s.y * w0x.y * w0x.y;
                *(v2f*)(b + eoff(2, f, nn)) = v;
                *(v2f*)(b + eoff(3, f, nn)) = at;
                *(v2f*)(b + eoff(4, f, nn)) = ax;
            }
        }
    }
    __syncthreads();

    // ---------------- 3 hidden layers: 256 -> 256 via WMMA f32 ----------
    int cur = 0;
    const float* Ws[3] = { W1, W2, W3 };
    const float* Bs[3] = { b1, b2, b3 };

    for (int L = 0; L < 3; ++L) {
        const float* __restrict__ W    = Ws[L];
        const float* __restrict__ bias = Bs[L];

        const float* in  = buf(cur);
        float*       o   = buf(cur ^ 1);
        // fragment base for k=0: eoff(0, 2*half, nn) with f=4k+2h
        const float* inb = in + half * NS * 32 + nn * 2;

        const int fb0 = sub * (16 / WPT);
        for (int fb = fb0; fb < fb0 + 16 / WPT; ++fb) {
            const int m_base = fb * 16;

            v8f acc[NS] = {};   // C = 0

#pragma unroll 4
            for (int k = 0; k < 64; ++k) {
                // A fragment: A[m, kk] = W[4k + 2*half + j][m_base + m]
                // (16 consecutive floats per half-wave; W is L2/L0 resident)
                const float* wp = W + (4 * k + 2 * half) * HID + m_base + nn;
                v2f a;
                a.x = wp[0];
                a.y = wp[HID];

                const float* kp = inb + k * (2 * NS * 32);
#pragma unroll
                for (int s2 = 0; s2 < NS; ++s2) {
                    // B fragment: one ds_load_b64, NS fragments 128B apart
                    v2f bfr = *(const v2f*)(kp + s2 * 32);
                    acc[s2] = __builtin_amdgcn_wmma_f32_16x16x4_f32(
                        false, a, false, bfr, (short)0, acc[s2], false, false);
                }
            }

            // elementwise tanh + tangent updates; paired b64 stores
#pragma unroll
            for (int jp = 0; jp < 4; ++jp) {
                const int j0 = 2 * jp;                 // C/D VGPR pair
                const int f0 = m_base + 8 * half + j0; // even feature index
                float* op = o + ((f0 >> 1) * NS) * 32 + nn * 2;
                v2f bb = *(const v2f*)(bias + f0);
                v2f h, s;
#pragma unroll
                for (int q = 0; q < 2; ++q) {
                    const float z = acc[0][j0 + q] + bb[q];
                    h[q] = tanhf(z);
                    s[q] = 1.0f - h[q] * h[q];
                }
                *(v2f*)(op) = h;
                if (NS >= 2) {
                    v2f zvt; zvt.x = acc[1][j0]; zvt.y = acc[1][j0 + 1];
                    v2f vt;  vt.x = s.x * zvt.x; vt.y = s.y * zvt.y;
                    *(v2f*)(op + 32) = vt;
                    if (NS == 5) {
                        v2f zvx; zvx.x = acc[2][j0]; zvx.y = acc[2][j0 + 1];
                        v2f vx;  vx.x = s.x * zvx.x; vx.y = s.y * zvx.y;
                        v2f at, ax;
                        at.x = s.x * acc[3][j0]     - 2.0f * h.x * s.x * zvt.x * zvt.x;
                        at.y = s.y * acc[3][j0 + 1] - 2.0f * h.y * s.y * zvt.y * zvt.y;
                        ax.x = s.x * acc[4][j0]     - 2.0f * h.x * s.x * zvx.x * zvx.x;
                        ax.y = s.y * acc[4][j0 + 1] - 2.0f * h.y * s.y * zvx.y * zvx.y;
                        *(v2f*)(op + 2 * 32) = vx;
                        *(v2f*)(op + 3 * 32) = at;
                        *(v2f*)(op + 4 * 32) = ax;
                    }
                }
            }
        }
        __syncthreads();
        cur ^= 1;
    }

    // ---------------- output layer: 256 -> 1 (dot-product reduce) --------
    if (sub == 0) {
        const float* fin = buf(cur);
        const int fbeg = half * (HID / 2);
        if (PHASE == 0) {
            // pde_residual = (a_t - a_x) @ W4     (C_WAVE = 1)
            float r = 0.0f;
            for (int f = fbeg; f < fbeg + HID / 2; f += 2) {
                v2f at = *(const v2f*)(fin + eoff(3, f, nn));
                v2f ax = *(const v2f*)(fin + eoff(4, f, nn));
                v2f w4 = *(const v2f*)(W4 + f);
                r += (at.x - ax.x) * w4.x + (at.y - ax.y) * w4.y;
            }
            r += __shfl_xor(r, 16);
            if (half == 0) out[0 * NPTS + tile * TILE + nn] = r;
        } else if (PHASE == 1) {
            // u_phi = h @ W4 + b4 ;  du_dt_psi = v_t @ W4
            float su = 0.0f, sv = 0.0f;
            for (int f = fbeg; f < fbeg + HID / 2; f += 2) {
                v2f hh = *(const v2f*)(fin + eoff(0, f, nn));
                v2f vv = *(const v2f*)(fin + eoff(1, f, nn));
                v2f w4 = *(const v2f*)(W4 + f);
                su += hh.x * w4.x + hh.y * w4.y;
                sv += vv.x * w4.x + vv.y * w4.y;
            }
            su += __shfl_xor(su, 16);
            sv += __shfl_xor(sv, 16);
            if (half == 0) {
                out[1 * NPTS + tile * TILE + nn] = su + b4[0];
                out[2 * NPTS + tile * TILE + nn] = sv;
            }
        } else {
            // u_bound = h @ W4 + b4
            float su = 0.0f;
            for (int f = fbeg; f < fbeg + HID / 2; f += 2) {
                v2f hh = *(const v2f*)(fin + eoff(0, f, nn));
                v2f w4 = *(const v2f*)(W4 + f);
                su += hh.x * w4.x + hh.y * w4.y;
            }
            su += __shfl_xor(su, 16);
            if (half == 0) out[3 * NPTS + tile * TILE + nn] = su + b4[0];
        }
    }
}

extern "C" void kernel_launch(void* const* d_in, const int* in_sizes, int n_in,
                              void* d_out, int out_size, void* d_ws, size_t ws_size,
                              hipStream_t stream) {
    (void)in_sizes; (void)n_in; (void)d_ws; (void)ws_size; (void)out_size;
    const float* W0 = (const float*)d_in[0];
    const float* b0 = (const float*)d_in[1];
    const float* W1 = (const float*)d_in[2];
    const float* b1 = (const float*)d_in[3];
    const float* W2 = (const float*)d_in[4];
    const float* b2 = (const float*)d_in[5];
    const float* W3 = (const float*)d_in[6];
    const float* b3 = (const float*)d_in[7];
    const float* W4 = (const float*)d_in[8];
    const float* b4 = (const float*)d_in[9];
    const float* tx_eq    = (const float*)d_in[10];
    const float* tx_init  = (const float*)d_in[11];
    const float* tx_bound = (const float*)d_in[12];
    float* out = (float*)d_out;

    dim3 blk(256);
    // equation: 5 streams, 2 tiles/WG (320KB LDS), 4 waves per tile
    pinn_kernel<5, 2, 4, 0><<<NTILES / 2, blk, 0, stream>>>(
        W0, b0, W1, b1, W2, b2, W3, b3, W4, b4, tx_eq, out);
    // init: 2 streams, 4 tiles/WG (256KB LDS), 2 waves per tile
    pinn_kernel<2, 4, 2, 1><<<NTILES / 4, blk, 0, stream>>>(
        W0, b0, W1, b1, W2, b2, W3, b3, W4, b4, tx_init, out);
    // boundary: 1 stream, 8 tiles/WG (256KB LDS), 1 wave per tile
    pinn_kernel<1, 8, 1, 2><<<NTILES / 8, blk, 0, stream>>>(
        W0, b0, W1, b1, W2, b2, W3, b3, W4, b4, tx_bound, out);
}